// Hybridization_PyTorch_model_39968965656937
// MI455X (gfx1250) — compile-verified
//
#include <hip/hip_runtime.h>
#include <cstdint>

// Problem constants (from the reference)
#define N_ITEMS 20000
#define N_REC   8
#define N_USERS 2048
#define U4      (N_USERS / 4)   // 512 float4 columns per row
#define TPB     128             // pass-1 threads/block (4 waves), covers 512 users
#define IPB     50              // items per block strip -> grid.y = 400
#define GRID_X  (U4 / TPB)      // 4
#define GRID_Y  (N_ITEMS / IPB) // 400

// ---------- order-preserving float<->uint mapping (total order incl. sign) ----------
__device__ __forceinline__ unsigned order_key(float f) {
  unsigned u = __float_as_uint(f);
  return u ^ ((unsigned)((int)u >> 31) | 0x80000000u);
}
__device__ __forceinline__ float key_to_float(unsigned k) {
  return __uint_as_float(k ^ ((unsigned)(~((int)k >> 31)) | 0x80000000u));
}

// ---------- CDNA5 async global->LDS copy (per-lane B128), non-temporal ----------
// lds_byte_off: workgroup-relative LDS byte address (truncated flat pointer).
// gaddr: 64-bit global address. NT hint keeps the 1.31GB stream out of L2 so the
// 164MB score matrix written by this pass stays resident for the normalize pass.
__device__ __forceinline__ void async_copy_b128_nt(uint32_t lds_byte_off,
                                                   unsigned long long gaddr) {
  asm volatile("global_load_async_to_lds_b128 %0, %1, off th:TH_LOAD_NT"
               :: "v"(lds_byte_off), "v"(gaddr)
               : "memory");
}
__device__ __forceinline__ void wait_async_le8() {
  asm volatile("s_wait_asynccnt 0x8" ::: "memory");
}
__device__ __forceinline__ void wait_async_le0() {
  asm volatile("s_wait_asynccnt 0x0" ::: "memory");
}

// ---------- kernel 1 (placed first so disasm snippet shows the async pipeline) ----
// Fused weighted-sum + per-user running max.
// Double-buffered async pipeline: 8 rows (R-slices) of one item per stage.
// LDS layout [buf][r][thread]: per r-slice, lanes are contiguous 16B -> no bank
// conflicts on either the async writes or the ds_load_b128 reads.
__global__ void __launch_bounds__(TPB) score_max_kernel(
    const float* __restrict__ in, const float* __restrict__ w,
    float* __restrict__ out, unsigned* __restrict__ maxkey)
{
  __shared__ float4 smem[2][N_REC][TPB];

  const int t     = threadIdx.x;
  const int c4    = blockIdx.x * TPB + t;    // float4 column index, 0..511
  const int item0 = blockIdx.y * IPB;

  float wr[N_REC];
#pragma unroll
  for (int r = 0; r < N_REC; ++r) wr[r] = w[r];

  const float4* in4  = (const float4*)in;
  float4*       out4 = (float4*)out;

  // Workgroup-relative LDS byte offsets for this lane's staging slots.
  // (Low 32 bits of a flat LDS pointer == LDS byte offset; this also makes
  //  smem "escape" so the asm memory clobbers cover it.)
  uint32_t lds_off[2][N_REC];
#pragma unroll
  for (int b = 0; b < 2; ++b)
#pragma unroll
    for (int r = 0; r < N_REC; ++r)
      lds_off[b][r] = (uint32_t)(uintptr_t)&smem[b][r][t];

  // Prologue: stream item0's 8 rows into buffer 0.
#pragma unroll
  for (int r = 0; r < N_REC; ++r)
    async_copy_b128_nt(lds_off[0][r],
        (unsigned long long)(uintptr_t)(in4 + (size_t)(item0 * N_REC + r) * U4 + c4));

  float4 mx = make_float4(-3.402823466e38f, -3.402823466e38f,
                          -3.402823466e38f, -3.402823466e38f);

  for (int it = 0; it < IPB; ++it) {
    const int b = it & 1;
    if (it + 1 < IPB) {
      const int item = item0 + it + 1;
#pragma unroll
      for (int r = 0; r < N_REC; ++r)
        async_copy_b128_nt(lds_off[b ^ 1][r],
            (unsigned long long)(uintptr_t)(in4 + (size_t)(item * N_REC + r) * U4 + c4));
      wait_async_le8();   // 16 outstanding; in-order completion -> current 8 landed
    } else {
      wait_async_le0();   // drain for the last item
    }

    float4 s = make_float4(0.f, 0.f, 0.f, 0.f);
#pragma unroll
    for (int r = 0; r < N_REC; ++r) {
      float4 x = smem[b][r][t];
      s.x = fmaf(wr[r], x.x, s.x);
      s.y = fmaf(wr[r], x.y, s.y);
      s.z = fmaf(wr[r], x.z, s.z);
      s.w = fmaf(wr[r], x.w, s.w);
    }

    out4[(size_t)(item0 + it) * U4 + c4] = s;

    mx.x = fmaxf(mx.x, s.x);
    mx.y = fmaxf(mx.y, s.y);
    mx.z = fmaxf(mx.z, s.z);
    mx.w = fmaxf(mx.w, s.w);
  }

  // One atomic per user per block strip (~820K atomics over 2048 addresses).
  const int u0 = c4 * 4;
  atomicMax(&maxkey[u0 + 0], order_key(mx.x));
  atomicMax(&maxkey[u0 + 1], order_key(mx.y));
  atomicMax(&maxkey[u0 + 2], order_key(mx.z));
  atomicMax(&maxkey[u0 + 3], order_key(mx.w));
}

// ---------- kernel 0: reset per-user max keys (0u < order_key of any real float) ----
__global__ void __launch_bounds__(256) init_keys_kernel(unsigned* __restrict__ keys) {
  keys[blockIdx.x * blockDim.x + threadIdx.x] = 0u;
}

// ---------- kernel 2: per-user reciprocal (2048 exact f32 divides) ----------------
__global__ void __launch_bounds__(256) finalize_inv_kernel(
    const unsigned* __restrict__ keys, float* __restrict__ rinv) {
  const int u = blockIdx.x * blockDim.x + threadIdx.x;
  rinv[u] = 1.0f / key_to_float(keys[u]);
}

// ---------- kernel 3: normalize (scores are hot in the 192MB L2) ------------------
__global__ void __launch_bounds__(256) normalize_kernel(
    float* __restrict__ out, const float* __restrict__ rinv)
{
  const size_t i4 = (size_t)blockIdx.x * blockDim.x + threadIdx.x; // float4 index
  float4* o4 = (float4*)out;
  float4 v = o4[i4];
  const float4 rv = ((const float4*)rinv)[i4 & (U4 - 1)];
  v.x *= rv.x; v.y *= rv.y; v.z *= rv.z; v.w *= rv.w;
  o4[i4] = v;
}

extern "C" void kernel_launch(void* const* d_in, const int* in_sizes, int n_in,
                              void* d_out, int out_size, void* d_ws, size_t ws_size,
                              hipStream_t stream) {
  const float* in = (const float*)d_in[0];   // [160000, 2048] f32
  const float* w  = (const float*)d_in[1];   // [1, 8] f32
  float* out = (float*)d_out;                // [20000, 2048] f32

  // ws layout: [0, 2048) uint max-keys | [2048, 4096) float reciprocals (16 KB)
  unsigned* keys = (unsigned*)d_ws;
  float*    rinv = (float*)((char*)d_ws + N_USERS * sizeof(unsigned));

  init_keys_kernel<<<N_USERS / 256, 256, 0, stream>>>(keys);

  dim3 g1(GRID_X, GRID_Y);
  score_max_kernel<<<g1, TPB, 0, stream>>>(in, w, out, keys);

  finalize_inv_kernel<<<N_USERS / 256, 256, 0, stream>>>(keys, rinv);

  normalize_kernel<<<(size_t)N_ITEMS * U4 / 256, 256, 0, stream>>>(out, rinv);
}